// GAT_13907104104868
// MI455X (gfx1250) — compile-verified
//
#include <hip/hip_runtime.h>
#include <math.h>

typedef float v2f __attribute__((ext_vector_type(2)));
typedef float v8f __attribute__((ext_vector_type(8)));

#define IN_DIM 128
#define HID    64
#define OUTD   64

// ---- order-preserving float <-> u32 encoding for atomicMax-based segment max ----
__device__ __forceinline__ unsigned enc_f32(float f) {
  unsigned b = __float_as_uint(f);
  return (b & 0x80000000u) ? ~b : (b | 0x80000000u);
}
__device__ __forceinline__ float dec_f32(unsigned u) {
  unsigned b = (u & 0x80000000u) ? (u & 0x7FFFFFFFu) : ~u;
  return __uint_as_float(b);
}

// ---------------- init: zero h (output accumulator), init mbits to enc(-inf), denom to 0 ----
__global__ void gat_init(float* __restrict__ h, unsigned* __restrict__ mbits,
                         float* __restrict__ denom, int N, long total) {
  long i = (long)blockIdx.x * blockDim.x + threadIdx.x;
  if (i < total) h[i] = 0.0f;
  if (i < N) {
    mbits[i] = enc_f32(-INFINITY);
    denom[i] = 0.0f;
  }
}

// ---------------- fused  z = gelu(x @ w_in + b_in) @ w  via V_WMMA_F32_16X16X4_F32 ----------
// One wave per 16-row tile; 2 waves (64 threads) per block.
__global__ __launch_bounds__(64) void gat_fused_gemm(
    const float* __restrict__ x, const float* __restrict__ w_in,
    const float* __restrict__ b_in, const float* __restrict__ w,
    float* __restrict__ z, int N) {
  __shared__ float lds[2][16][66];  // per-wave 16x64 h0 tile, padded to dodge bank conflicts

  const int lane = threadIdx.x & 31;
  const int wv   = threadIdx.x >> 5;
  const int tile = blockIdx.x * 2 + wv;
  const int ntiles = (N + 15) >> 4;
  const bool active = tile < ntiles;

  const int m  = lane & 15;   // A row within tile / B,C,D column within tile
  const int kp = lane >> 4;   // which K pair this half-wave holds

  if (active) {
    const int row  = tile * 16 + m;
    const int rowc = row < N ? row : (N - 1);  // clamp for tail tiles; stores are guarded
    const float* xrow = x + (long)rowc * IN_DIM;

    v8f acc[4];
    #pragma unroll
    for (int nt = 0; nt < 4; ++nt) acc[nt] = (v8f){0,0,0,0,0,0,0,0};

    // ---- GEMM1: h0 = x(16x128) @ w_in(128x64), K stepped by 4 ----
    for (int k0 = 0; k0 < IN_DIM; k0 += 4) {
      const int kk = k0 + 2 * kp;
      v2f av = *(const v2f*)(xrow + kk);            // A: {x[m][kk], x[m][kk+1]}
      const float* wr0 = w_in + (long)kk * HID;     // B rows kk, kk+1
      const float* wr1 = wr0 + HID;
      #pragma unroll
      for (int nt = 0; nt < 4; ++nt) {
        v2f bv = { wr0[nt * 16 + m], wr1[nt * 16 + m] };
        acc[nt] = __builtin_amdgcn_wmma_f32_16x16x4_f32(
            false, av, false, bv, (short)0, acc[nt], false, false);
      }
    }

    // ---- bias + exact GELU, re-layout C->A via LDS ----
    #pragma unroll
    for (int nt = 0; nt < 4; ++nt) {
      float bias = b_in[nt * 16 + m];
      #pragma unroll
      for (int r = 0; r < 8; ++r) {
        float v = acc[nt][r] + bias;
        v = 0.5f * v * (1.0f + erff(v * 0.70710678118654752f));  // exact GELU
        lds[wv][r + 8 * kp][nt * 16 + m] = v;
      }
    }
  }
  __syncthreads();

  if (active) {
    v8f acc[4];
    #pragma unroll
    for (int nt = 0; nt < 4; ++nt) acc[nt] = (v8f){0,0,0,0,0,0,0,0};

    // ---- GEMM2: z = h0(16x64) @ w(64x64) ----
    for (int k0 = 0; k0 < HID; k0 += 4) {
      const int kk = k0 + 2 * kp;
      v2f av = *(const v2f*)(&lds[wv][m][kk]);
      const float* wr0 = w + (long)kk * OUTD;
      const float* wr1 = wr0 + OUTD;
      #pragma unroll
      for (int nt = 0; nt < 4; ++nt) {
        v2f bv = { wr0[nt * 16 + m], wr1[nt * 16 + m] };
        acc[nt] = __builtin_amdgcn_wmma_f32_16x16x4_f32(
            false, av, false, bv, (short)0, acc[nt], false, false);
      }
    }

    // ---- store z tile ----
    #pragma unroll
    for (int nt = 0; nt < 4; ++nt) {
      #pragma unroll
      for (int r = 0; r < 8; ++r) {
        int rr = tile * 16 + r + 8 * kp;
        if (rr < N) z[(long)rr * OUTD + nt * 16 + m] = acc[nt][r];
      }
    }
  }
}

// ---------------- per-node attention scalars: s1 = z . a[:64], s2 = z . a[64:] -------------
__global__ __launch_bounds__(256) void gat_scores(
    const float* __restrict__ z, const float* __restrict__ a,
    float* __restrict__ s1, float* __restrict__ s2, int N) {
  int i = blockIdx.x * blockDim.x + threadIdx.x;
  if (i >= N) return;
  const float4* zr = (const float4*)(z + (long)i * OUTD);
  const float4* a1 = (const float4*)a;
  const float4* a2 = (const float4*)(a + OUTD);
  float acc1 = 0.f, acc2 = 0.f;
  #pragma unroll
  for (int j = 0; j < OUTD / 4; ++j) {
    float4 v = zr[j];
    float4 p = a1[j];
    float4 q = a2[j];
    acc1 += v.x * p.x + v.y * p.y + v.z * p.z + v.w * p.w;
    acc2 += v.x * q.x + v.y * q.y + v.z * q.z + v.w * q.w;
  }
  s1[i] = acc1;
  s2[i] = acc2;
}

// ---------------- pass 1 over edges: e = leaky_relu(s1[src]+s2[dst]); segment max ----------
__global__ __launch_bounds__(256) void gat_edge_max(
    const int* __restrict__ src, const int* __restrict__ dst,
    const float* __restrict__ s1, const float* __restrict__ s2,
    float* __restrict__ ebuf, unsigned* __restrict__ mbits, int E) {
  int e = blockIdx.x * blockDim.x + threadIdx.x;
  if (e >= E) return;
  int s = src[e], d = dst[e];
  float v = s1[s] + s2[d];
  v = v > 0.f ? v : 0.01f * v;   // leaky_relu
  ebuf[e] = v;
  atomicMax(&mbits[d], enc_f32(v));
}

// ---------------- pass 2: ex = exp(e - m[dst]); segment sum of ex --------------------------
__global__ __launch_bounds__(256) void gat_edge_exp(
    const int* __restrict__ dst, float* __restrict__ ebuf,
    const unsigned* __restrict__ mbits, float* __restrict__ denom, int E) {
  int e = blockIdx.x * blockDim.x + threadIdx.x;
  if (e >= E) return;
  int d = dst[e];
  float mv = dec_f32(mbits[d]);
  if (!isfinite(mv)) mv = 0.0f;
  float ev = expf(ebuf[e] - mv);
  ebuf[e] = ev;
  atomicAdd(&denom[d], ev);
}

// ---------------- pass 3: h[dst] += (ex/denom[dst]) * z[src]; one wave32 per edge ----------
__global__ __launch_bounds__(256) void gat_scatter(
    const int* __restrict__ src, const int* __restrict__ dst,
    const float* __restrict__ z, const float* __restrict__ ebuf,
    const float* __restrict__ denom, float* __restrict__ h, int E) {
  int e = blockIdx.x * 8 + (threadIdx.x >> 5);
  if (e >= E) return;
  int lane = threadIdx.x & 31;
  int s = src[e], d = dst[e];
  float alpha = ebuf[e] / fmaxf(denom[d], 1e-9f);
  v2f zv = *(const v2f*)(z + (long)s * OUTD + 2 * lane);
  float* hp = h + (long)d * OUTD + 2 * lane;
  atomicAdd(hp + 0, alpha * zv.x);
  atomicAdd(hp + 1, alpha * zv.y);
}

// ------------------------------------ launcher --------------------------------------------
extern "C" void kernel_launch(void* const* d_in, const int* in_sizes, int n_in,
                              void* d_out, int out_size, void* d_ws, size_t ws_size,
                              hipStream_t stream) {
  const float* x    = (const float*)d_in[0];
  const float* w_in = (const float*)d_in[1];
  const float* b_in = (const float*)d_in[2];
  const float* w    = (const float*)d_in[3];
  const float* a    = (const float*)d_in[4];
  const int*   src  = (const int*)d_in[5];
  const int*   dst  = (const int*)d_in[6];
  float* h = (float*)d_out;

  const int N = in_sizes[0] / IN_DIM;
  const int E = in_sizes[5];

  // workspace layout (256B aligned slices)
  auto align256 = [](size_t v) { return (v + 255) & ~(size_t)255; };
  char* base = (char*)d_ws;
  size_t off = 0;
  float*    z     = (float*)(base + off); off += align256((size_t)N * OUTD * 4);
  float*    s1    = (float*)(base + off); off += align256((size_t)N * 4);
  float*    s2    = (float*)(base + off); off += align256((size_t)N * 4);
  unsigned* mbits = (unsigned*)(base + off); off += align256((size_t)N * 4);
  float*    denom = (float*)(base + off); off += align256((size_t)N * 4);
  float*    ebuf  = (float*)(base + off); off += align256((size_t)E * 4);

  const long total = (long)N * OUTD;
  {
    long blocks = (total + 255) / 256;
    gat_init<<<dim3((unsigned)blocks), dim3(256), 0, stream>>>(h, mbits, denom, N, total);
  }
  {
    int ntiles = (N + 15) >> 4;
    int blocks = (ntiles + 1) / 2;   // 2 waves per block
    gat_fused_gemm<<<dim3(blocks), dim3(64), 0, stream>>>(x, w_in, b_in, w, z, N);
  }
  gat_scores<<<dim3((N + 255) / 256), dim3(256), 0, stream>>>(z, a, s1, s2, N);
  gat_edge_max<<<dim3((E + 255) / 256), dim3(256), 0, stream>>>(src, dst, s1, s2, ebuf, mbits, E);
  gat_edge_exp<<<dim3((E + 255) / 256), dim3(256), 0, stream>>>(dst, ebuf, mbits, denom, E);
  gat_scatter<<<dim3((E + 7) / 8), dim3(256), 0, stream>>>(src, dst, z, ebuf, denom, h, E);
}